// UnionRGATLayer_12180527251907
// MI455X (gfx1250) — compile-verified
//
#include <hip/hip_runtime.h>
#include <hip/hip_bf16.h>
#include <math.h>

#define FD 64  // feature dim

typedef float v2f __attribute__((ext_vector_type(2)));
typedef float v8f __attribute__((ext_vector_type(8)));

// ---------------------------------------------------------------------------
// v = attn_w1 @ attn_w2   (192x64 @ 64x1 -> 192)
// ---------------------------------------------------------------------------
__global__ void k_compute_v(const float* __restrict__ W1,
                            const float* __restrict__ w2,
                            float* __restrict__ v) {
    __shared__ float w2s[FD];
    if (threadIdx.x < FD) w2s[threadIdx.x] = w2[threadIdx.x];
    __syncthreads();
    int i = threadIdx.x;
    if (i < 3 * FD) {
        float acc = 0.f;
        #pragma unroll 8
        for (int j = 0; j < FD; ++j) acc += W1[i * FD + j] * w2s[j];
        v[i] = acc;
    }
}

// ---------------------------------------------------------------------------
// per-node scores: s1[n] = h[n].v[0:64], s2[n] = h[n].v[64:128]
// ---------------------------------------------------------------------------
__global__ void k_node_score(const float* __restrict__ h,
                             const float* __restrict__ v,
                             float* __restrict__ s1,
                             float* __restrict__ s2, int N) {
    __shared__ float vs[2 * FD];
    if (threadIdx.x < 2 * FD) vs[threadIdx.x] = v[threadIdx.x];
    __syncthreads();
    int n = blockIdx.x * blockDim.x + threadIdx.x;
    if (n >= N) return;
    const float4* hr = (const float4*)(h + (size_t)n * FD);
    float a = 0.f, b = 0.f;
    #pragma unroll
    for (int j = 0; j < FD / 4; ++j) {
        float4 x = hr[j];
        a += x.x * vs[4 * j + 0] + x.y * vs[4 * j + 1] +
             x.z * vs[4 * j + 2] + x.w * vs[4 * j + 3];
        b += x.x * vs[FD + 4 * j + 0] + x.y * vs[FD + 4 * j + 1] +
             x.z * vs[FD + 4 * j + 2] + x.w * vs[FD + 4 * j + 3];
    }
    s1[n] = a;
    s2[n] = b;
}

// ---------------------------------------------------------------------------
// per-relation scores: sr[t] = emb_rel[t] . v[128:192]
// ---------------------------------------------------------------------------
__global__ void k_rel_score(const float* __restrict__ emb_rel,
                            const float* __restrict__ v,
                            float* __restrict__ sr, int R) {
    __shared__ float v3[FD];
    if (threadIdx.x < FD) v3[threadIdx.x] = v[2 * FD + threadIdx.x];
    __syncthreads();
    int t = blockIdx.x * blockDim.x + threadIdx.x;
    if (t >= R) return;
    float acc = 0.f;
    #pragma unroll 8
    for (int j = 0; j < FD; ++j) acc += emb_rel[t * FD + j] * v3[j];
    sr[t] = acc;
}

// ---------------------------------------------------------------------------
// self-loop GEMM via V_WMMA_F32_16X16X4_F32:  loop = h @ loop_weight
// block = 256 threads = 8 waves; each wave computes a 16x64 output tile.
//
// A fragment (16x4 f32, 2 VGPR): lane L holds A[M=L&15][K=4k+2*(L>>4) .. +1]
//   -> one 8B-aligned global b64 load per k-step, preloaded before the loop.
// B staged in LDS as interleaved float2: Bsh[j][n] = {W[2j][n], W[2j+1][n]}
//   -> one ds_load_b64 per fragment. Pitch BP=80 float2 puts the two wave
//      halves (j vs j+1) on disjoint 32-bank sets (2*80 mod 64 == 32).
// C/D (16x16 f32, 8 VGPR): lanes 0-15 M=r, lanes 16-31 M=r+8, N=lane&15.
// Out-of-range rows are clamped (not predicated) on load; only the store is
// guarded, so the WMMA loop is branch-free with full EXEC.
// ---------------------------------------------------------------------------
#define BP 80  // Bsh pitch in float2
__global__ __launch_bounds__(256) void k_loop_gemm(const float* __restrict__ h,
                                                   const float* __restrict__ W,
                                                   float* __restrict__ loop,
                                                   int N) {
    __shared__ v2f Bsh[32 * BP];
    for (int t = threadIdx.x; t < 32 * FD; t += 256) {
        int j = t >> 6, n = t & 63;
        v2f b;
        b.x = W[(2 * j) * FD + n];
        b.y = W[(2 * j + 1) * FD + n];
        Bsh[j * BP + n] = b;
    }
    __syncthreads();

    const int lane    = threadIdx.x & 31;
    const int lrow    = lane & 15;
    const int khalf   = lane >> 4;
    const int rowBase = blockIdx.x * 128 + (threadIdx.x >> 5) * 16;

    const int myRow = rowBase + lrow;
    const float* hrow = h + (size_t)(myRow < N ? myRow : 0) * FD;

    v2f afrag[16];
    #pragma unroll
    for (int k = 0; k < 16; ++k)
        afrag[k] = *(const v2f*)(hrow + 4 * k + 2 * khalf);

    v8f acc[4] = {};
    #pragma unroll
    for (int k = 0; k < 16; ++k) {
        const int jj = 2 * k + khalf;
        #pragma unroll
        for (int nt = 0; nt < 4; ++nt) {
            v2f b = Bsh[jj * BP + nt * 16 + lrow];
            acc[nt] = __builtin_amdgcn_wmma_f32_16x16x4_f32(
                false, afrag[k], false, b, (short)0, acc[nt], false, false);
        }
    }

    #pragma unroll
    for (int nt = 0; nt < 4; ++nt) {
        #pragma unroll
        for (int r = 0; r < 8; ++r) {
            int row = rowBase + r + 8 * khalf;
            if (row < N) loop[(size_t)row * FD + nt * 16 + lrow] = acc[nt][r];
        }
    }
}

// ---------------------------------------------------------------------------
// init:  m = -inf, denom = 0, agg = 0
// ---------------------------------------------------------------------------
__global__ void k_init(float* __restrict__ m, float* __restrict__ denom,
                       float* __restrict__ agg, int N) {
    size_t idx = (size_t)blockIdx.x * blockDim.x + threadIdx.x;
    if (idx < (size_t)N * FD) agg[idx] = 0.f;
    if (idx < (size_t)N) {
        m[idx]     = -INFINITY;
        denom[idx] = 0.f;
    }
}

// float atomic-max via sign-aware int min/max (init value is -inf)
__device__ __forceinline__ void atomicMaxF(float* addr, float val) {
    if (val >= 0.f)
        atomicMax((int*)addr, __float_as_int(val));
    else
        atomicMin((unsigned int*)addr, __float_as_uint(val));
}

// ---------------------------------------------------------------------------
// e[i] = leaky_relu(s1[src]+s2[dst]+sr[etype]);  m[dst] = max(m[dst], e)
// ---------------------------------------------------------------------------
__global__ void k_edge_score(const int* __restrict__ src,
                             const int* __restrict__ dst,
                             const int* __restrict__ et,
                             const float* __restrict__ s1,
                             const float* __restrict__ s2,
                             const float* __restrict__ sr,
                             float* __restrict__ e,
                             float* __restrict__ m, long long E) {
    long long i = (long long)blockIdx.x * blockDim.x + threadIdx.x;
    if (i >= E) return;
    int d = dst[i];
    float x = s1[src[i]] + s2[d] + sr[et[i]];
    x = (x > 0.f) ? x : 0.01f * x;
    e[i] = x;
    atomicMaxF(&m[d], x);
}

// ---------------------------------------------------------------------------
// e[i] <- exp(e[i]-m[dst]);  denom[dst] += e[i]
// ---------------------------------------------------------------------------
__global__ void k_edge_exp(const int* __restrict__ dst,
                           float* __restrict__ e,
                           const float* __restrict__ m,
                           float* __restrict__ denom, long long E) {
    long long i = (long long)blockIdx.x * blockDim.x + threadIdx.x;
    if (i >= E) return;
    int d = dst[i];
    float ex = expf(e[i] - m[d]);
    e[i] = ex;
    unsafeAtomicAdd(&denom[d], ex);
}

// ---------------------------------------------------------------------------
// agg[dst, j] += (e[i]/denom[dst]) * h[src, j]   (one thread per edge-feature)
// ---------------------------------------------------------------------------
__global__ void k_edge_agg(const int* __restrict__ src,
                           const int* __restrict__ dst,
                           const float* __restrict__ e,
                           const float* __restrict__ denom,
                           const float* __restrict__ h,
                           float* __restrict__ agg, long long E) {
    long long t = (long long)blockIdx.x * blockDim.x + threadIdx.x;
    long long i = t >> 6;
    int j = (int)(t & 63);
    if (i >= E) return;
    int d = dst[i], s = src[i];
    float w = e[i] / denom[d];
    unsafeAtomicAdd(&agg[(size_t)d * FD + j], w * h[(size_t)s * FD + j]);
}

// ---------------------------------------------------------------------------
// out = relu( (deg>0 ? agg : h) + loop )   ; deg>0 <=> m != -inf
// ---------------------------------------------------------------------------
__global__ void k_finalize(const float* __restrict__ agg,
                           const float* __restrict__ h,
                           const float* __restrict__ loop,
                           const float* __restrict__ m,
                           float* __restrict__ out, int N) {
    size_t idx = (size_t)blockIdx.x * blockDim.x + threadIdx.x;
    if (idx >= (size_t)N * FD) return;
    int n = (int)(idx >> 6);
    float base = (m[n] > -INFINITY) ? agg[idx] : h[idx];
    float y = base + loop[idx];
    out[idx] = (y > 0.f) ? y : 0.f;
}

// ---------------------------------------------------------------------------
extern "C" void kernel_launch(void* const* d_in, const int* in_sizes, int n_in,
                              void* d_out, int out_size, void* d_ws, size_t ws_size,
                              hipStream_t stream) {
    const float* h     = (const float*)d_in[0];
    const float* rel   = (const float*)d_in[1];
    const float* loopW = (const float*)d_in[2];
    const float* W1    = (const float*)d_in[3];
    const float* w2    = (const float*)d_in[4];
    const int*   src   = (const int*)d_in[5];
    const int*   dst   = (const int*)d_in[6];
    const int*   et    = (const int*)d_in[7];
    float* out = (float*)d_out;

    const int       N = in_sizes[0] / FD;
    const int       R = in_sizes[1] / FD;
    const long long E = in_sizes[5];

    // workspace layout (floats)
    float* ws    = (float*)d_ws;
    float* loop  = ws;                        // N*FD
    float* agg   = loop  + (size_t)N * FD;    // N*FD
    float* m     = agg   + (size_t)N * FD;    // N
    float* denom = m     + N;                 // N
    float* s1    = denom + N;                 // N
    float* s2    = s1    + N;                 // N
    float* e     = s2    + N;                 // E
    float* v     = e     + E;                 // 192
    float* sr    = v     + 192;               // R

    // 1. v = W1 @ w2
    k_compute_v<<<1, 256, 0, stream>>>(W1, w2, v);
    // 2. per-node and per-relation attention scores
    k_node_score<<<(N + 255) / 256, 256, 0, stream>>>(h, v, s1, s2, N);
    k_rel_score<<<(R + 255) / 256, 256, 0, stream>>>(rel, v, sr, R);
    // 3. self-loop GEMM (WMMA f32)
    k_loop_gemm<<<(N + 127) / 128, 256, 0, stream>>>(h, loopW, loop, N);
    // 4. init segment buffers
    {
        size_t tot = (size_t)N * FD;
        k_init<<<(unsigned)((tot + 255) / 256), 256, 0, stream>>>(m, denom, agg, N);
    }
    // 5. edge scores + segment max
    k_edge_score<<<(unsigned)((E + 255) / 256), 256, 0, stream>>>(
        src, dst, et, s1, s2, sr, e, m, E);
    // 6. exp + segment sum
    k_edge_exp<<<(unsigned)((E + 255) / 256), 256, 0, stream>>>(dst, e, m, denom, E);
    // 7. attention-weighted scatter aggregate
    {
        long long tot = E * FD;
        k_edge_agg<<<(unsigned)((tot + 255) / 256), 256, 0, stream>>>(
            src, dst, e, denom, h, agg, E);
    }
    // 8. finalize
    {
        size_t tot = (size_t)N * FD;
        k_finalize<<<(unsigned)((tot + 255) / 256), 256, 0, stream>>>(
            agg, h, loop, m, out, N);
    }
}